// BlockSelfAttention2_66606352826941
// MI455X (gfx1250) — compile-verified
//
#include <hip/hip_runtime.h>
#include <hip/hip_bf16.h>

typedef float v2f __attribute__((ext_vector_type(2)));
typedef float v8f __attribute__((ext_vector_type(8)));

#define HWDIM 504
#define HWTOT (504 * 504)
#define TS 68              // token stride in Y2 (padded from 64)
#define RS (16 * TS)       // per-channel stride = 1088 floats
#define WPS 68             // w_ptg row stride (48 rows x 68)
#define WOS 18             // w_out row stride (64 rows x 18)
#define SROW 18            // score row stride (16 rows x 18)
#define Y2_OFF 0           // 48*1088 = 52224
#define S_OFF 52224        // 16*18   = 288
#define WPT_OFF 52512      // 48*68   = 3264
#define BPT_OFF 55776      // 48
#define WOU_OFF 55824      // 64*18   = 1152
#define BOU_OFF 56976      // 64
#define LDS_FLOATS 57040   // 228160 bytes

// D = A(16x4 f32) x B(4x16 f32) + C(16x16 f32), full fp32 precision.
static __device__ __forceinline__ v8f wmma4(v2f a, v2f b, v8f c) {
  return __builtin_amdgcn_wmma_f32_16x16x4_f32(false, a, false, b, (short)0, c,
                                               false, false);
}
static __device__ __forceinline__ v2f ldsv2(const float* p) {
  return *reinterpret_cast<const v2f*>(p); // 8B-aligned by construction
}

__global__ __launch_bounds__(256) void fused_tile_attn(
    const float* __restrict__ x, const float* __restrict__ w_ptg,
    const float* __restrict__ b_ptg, const float* __restrict__ w_out,
    const float* __restrict__ b_out, float* __restrict__ out) {
  extern __shared__ float lds[];
  float* Y2 = lds + Y2_OFF;   // [48 ch][16 tok][64 pos] (padded)
  float* Sm = lds + S_OFF;    // 16x16 scores, row stride 18
  float* WPT = lds + WPT_OFF; // w_ptg row-major, row stride 68
  float* BPT = lds + BPT_OFF;
  float* WOU = lds + WOU_OFF; // w_out row-major, row stride 18
  float* BOU = lds + BOU_OFF;

  const int tid = threadIdx.x;
  const int lane = tid & 31;
  const int wv = tid >> 5;  // 8 waves
  const int n = lane & 15;  // M/N row index inside 16-wide tile
  const int hi = lane >> 4; // K upper-half selector

  const int blk = blockIdx.x; // 1024 = 4 batch * 16 * 16 tiles of 32x32 px
  const int b = blk >> 8;
  const int H2 = (blk >> 4) & 15;
  const int W2 = blk & 15;

  // ---- Phase 0: stage weights (padded row-major, coalesced), zero scores ----
  for (int i = tid; i < 48 * 64; i += 256) WPT[(i >> 6) * WPS + (i & 63)] = w_ptg[i];
  for (int i = tid; i < 64 * 16; i += 256) WOU[(i >> 4) * WOS + (i & 15)] = w_out[i];
  if (tid < 48) BPT[tid] = b_ptg[tid];
  if (tid < 64) BOU[tid] = b_out[tid];
  for (int i = tid; i < 16 * SROW; i += 256) Sm[i] = 0.0f;
  __syncthreads();

  const size_t xb = (size_t)b * 64 * HWTOT;

  // ---- Phase 1: Y = Wptg(48x64) * X(64x1024) + bias ; zero in pad fringe ----
  for (int nt = wv * 8; nt < wv * 8 + 8; ++nt) {
    const int p = nt * 16 + n;
    const int py = p >> 5, px = p & 31;
    const int h = H2 * 32 + py, w = W2 * 32 + px;
    const bool valid = (h < HWDIM) && (w < HWDIM);
    const float msk = valid ? 1.0f : 0.0f;
    // clamp -> always-in-bounds address, branchless masked value
    const int hc = h < HWDIM ? h : (HWDIM - 1);
    const int wc = w < HWDIM ? w : (HWDIM - 1);
    const size_t xoff = xb + (size_t)hc * HWDIM + wc;
    v8f acc0 = {}, acc1 = {}, acc2 = {};
    for (int k = 0; k < 64; k += 4) {
      const int cA = k + 2 * hi; // this lane supplies K rows {cA, cA+1}
      v2f bm;
      bm.x = x[xoff + (size_t)cA * HWTOT] * msk;
      bm.y = x[xoff + (size_t)(cA + 1) * HWTOT] * msk;
      v2f a0 = ldsv2(&WPT[n * WPS + cA]);
      v2f a1 = ldsv2(&WPT[(16 + n) * WPS + cA]);
      v2f a2 = ldsv2(&WPT[(32 + n) * WPS + cA]);
      acc0 = wmma4(a0, bm, acc0);
      acc1 = wmma4(a1, bm, acc1);
      acc2 = wmma4(a2, bm, acc2);
    }
    const int t = ((py >> 3) << 2) | (px >> 3);
    const int pos = ((py & 7) << 3) | (px & 7);
    const int lo = t * TS + pos;
#pragma unroll
    for (int v = 0; v < 8; ++v) {
      const int m0 = v + 8 * hi;
      Y2[(m0)*RS + lo]        = (acc0[v] + BPT[m0]) * msk;
      Y2[(m0 + 16) * RS + lo] = (acc1[v] + BPT[m0 + 16]) * msk;
      Y2[(m0 + 32) * RS + lo] = (acc2[v] + BPT[m0 + 32]) * msk;
    }
  }
  __syncthreads();

  // ---- Phase 2: S = Qn(16x1024) * Qt^T ; K split across 8 waves ----
  {
    v8f acc = {};
    const int kb = wv * 128;
    for (int kk = 0; kk < 32; ++kk) {
      const int k0 = kb + kk * 4;
      const int c = k0 >> 6;
      const int pos0 = (k0 & 63) + 2 * hi; // even -> 8B aligned
      v2f a = ldsv2(&Y2[c * RS + n * TS + pos0]);
      v2f bm = ldsv2(&Y2[(16 + c) * RS + n * TS + pos0]);
      acc = wmma4(a, bm, acc);
    }
#pragma unroll
    for (int v = 0; v < 8; ++v) atomicAdd(&Sm[(v + 8 * hi) * SROW + n], acc[v]);
  }
  __syncthreads();

  // ---- Phase 2b: row softmax over j ----
  if (tid < 16) {
    float row[16];
    float mx = -3.4e38f;
#pragma unroll
    for (int j = 0; j < 16; ++j) { row[j] = Sm[tid * SROW + j]; mx = fmaxf(mx, row[j]); }
    float sum = 0.0f;
#pragma unroll
    for (int j = 0; j < 16; ++j) { row[j] = expf(row[j] - mx); sum += row[j]; }
    const float inv = 1.0f / sum;
#pragma unroll
    for (int j = 0; j < 16; ++j) Sm[tid * SROW + j] = row[j] * inv;
  }
  __syncthreads();

  // ---- Phase 3: O = SA(16x16) * G(16x1024) ; overwrite Qn region of Y2 ----
  for (int nt = wv * 8; nt < wv * 8 + 8; ++nt) {
    const int f = nt * 16 + n;
    const int c1 = f >> 6;
    const int fo = f & 63;
    v8f acc = {};
#pragma unroll
    for (int k0 = 0; k0 < 16; k0 += 4) {
      v2f a = ldsv2(&Sm[n * SROW + k0 + 2 * hi]);
      v2f bm;
      bm.x = Y2[(32 + c1) * RS + (k0 + 2 * hi) * TS + fo];
      bm.y = Y2[(32 + c1) * RS + (k0 + 2 * hi + 1) * TS + fo];
      acc = wmma4(a, bm, acc);
    }
#pragma unroll
    for (int v = 0; v < 8; ++v) Y2[c1 * RS + (v + 8 * hi) * TS + fo] = acc[v];
  }
  __syncthreads();

  // ---- Phase 4: out = Wout(64x16) * O + bias + residual x ----
  for (int nt = wv * 8; nt < wv * 8 + 8; ++nt) {
    const int p = nt * 16 + n;
    const int py = p >> 5, px = p & 31;
    const int h = H2 * 32 + py, w = W2 * 32 + px;
    const bool valid = (h < HWDIM) && (w < HWDIM);
    const int t = ((py >> 3) << 2) | (px >> 3);
    const int pos = ((py & 7) << 3) | (px & 7);
    v8f acc0 = {}, acc1 = {}, acc2 = {}, acc3 = {};
#pragma unroll
    for (int k0 = 0; k0 < 16; k0 += 4) {
      const int c1 = k0 + 2 * hi;
      v2f bm;
      bm.x = Y2[c1 * RS + t * TS + pos];
      bm.y = Y2[(c1 + 1) * RS + t * TS + pos];
      v2f a0 = ldsv2(&WOU[n * WOS + c1]);
      v2f a1 = ldsv2(&WOU[(16 + n) * WOS + c1]);
      v2f a2 = ldsv2(&WOU[(32 + n) * WOS + c1]);
      v2f a3 = ldsv2(&WOU[(48 + n) * WOS + c1]);
      acc0 = wmma4(a0, bm, acc0);
      acc1 = wmma4(a1, bm, acc1);
      acc2 = wmma4(a2, bm, acc2);
      acc3 = wmma4(a3, bm, acc3);
    }
    if (valid) {
      const size_t gbase = xb + (size_t)h * HWDIM + w;
#pragma unroll
      for (int v = 0; v < 8; ++v) {
        const int m0 = v + 8 * hi;
        size_t g0 = gbase + (size_t)(m0)*HWTOT;
        size_t g1 = gbase + (size_t)(m0 + 16) * HWTOT;
        size_t g2 = gbase + (size_t)(m0 + 32) * HWTOT;
        size_t g3 = gbase + (size_t)(m0 + 48) * HWTOT;
        out[g0] = acc0[v] + BOU[m0] + x[g0];
        out[g1] = acc1[v] + BOU[m0 + 16] + x[g1];
        out[g2] = acc2[v] + BOU[m0 + 32] + x[g2];
        out[g3] = acc3[v] + BOU[m0 + 48] + x[g3];
      }
    }
  }
}

extern "C" void kernel_launch(void* const* d_in, const int* in_sizes, int n_in,
                              void* d_out, int out_size, void* d_ws,
                              size_t ws_size, hipStream_t stream) {
  const float* x = (const float*)d_in[0];
  const float* w_ptg = (const float*)d_in[1];
  const float* b_ptg = (const float*)d_in[2];
  const float* w_out = (const float*)d_in[3];
  const float* b_out = (const float*)d_in[4];
  float* out = (float*)d_out;

  const size_t shmem = (size_t)LDS_FLOATS * sizeof(float);
  (void)hipFuncSetAttribute((const void*)fused_tile_attn,
                            hipFuncAttributeMaxDynamicSharedMemorySize,
                            (int)shmem);
  fused_tile_attn<<<dim3(1024), dim3(256), shmem, stream>>>(x, w_ptg, b_ptg,
                                                            w_out, b_out, out);
}